// EM_GridSimulator_6803228196900
// MI455X (gfx1250) — compile-verified
//
#include <hip/hip_runtime.h>
#include <math.h>

// ---------------------------------------------------------------------------
// EM grid simulator: out[b] = exp(-sum_s grid_lookup(x[b,s]) * dist[b,s])
//   dist[b,0] = 1, dist[b,s] = 7*||x[b,s]-x[b,s-1]||
// Memory-bound: 50.3 MB of x streamed once (async global->LDS), 8 MB grid is
// L2-resident (192 MB L2), gathers served from L2. HBM floor ~2.2 us @ 23.3 TB/s.
// ---------------------------------------------------------------------------

#define NS               512
#define GRES             128
#define SCALE            7.0f
#define WAVES_PER_BLOCK  8
#define LANES            32
#define SAMPLES_PER_LANE (NS / LANES)          // 16
#define FLOATS_PER_RAY   (NS * 3)              // 1536 floats = 6144 B
#define VEC4_PER_RAY     (FLOATS_PER_RAY / 4)  // 384
#define VEC4_PER_LANE    (VEC4_PER_RAY / LANES)// 12

typedef int v4i __attribute__((__vector_size__(4 * sizeof(int))));
#define AS_GLOBAL __attribute__((address_space(1)))
#define AS_LDS    __attribute__((address_space(3)))

__global__ __launch_bounds__(WAVES_PER_BLOCK * LANES)
void em_grid_kernel(const float* __restrict__ x,
                    const float* __restrict__ grid,
                    float* __restrict__ out,
                    int n_rays)
{
    __shared__ float smem[WAVES_PER_BLOCK * FLOATS_PER_RAY]; // 48 KB / block

    const int lane = threadIdx.x & (LANES - 1);
    const int wave = threadIdx.x >> 5;
    const int ray  = blockIdx.x * WAVES_PER_BLOCK + wave;
    if (ray >= n_rays) return; // wave-uniform predicate

    const float* gx = x + (size_t)ray * FLOATS_PER_RAY;
    float*       lx = &smem[wave * FLOATS_PER_RAY];

    // ---- Stage this ray's coordinates into LDS (CDNA5 async copy path) ----
    #pragma unroll
    for (int k = 0; k < VEC4_PER_LANE; ++k) {
        const int v4 = k * LANES + lane; // 512B contiguous per instruction
#if __has_builtin(__builtin_amdgcn_global_load_async_to_lds_b128)
        __builtin_amdgcn_global_load_async_to_lds_b128(
            (AS_GLOBAL v4i*)(gx + v4 * 4),
            (AS_LDS    v4i*)(lx + v4 * 4),
            /*imm offset*/ 0, /*cpol*/ 0);
#else
        ((float4*)lx)[v4] = ((const float4*)gx)[v4];
#endif
    }
#if __has_builtin(__builtin_amdgcn_s_wait_asynccnt)
    __builtin_amdgcn_s_wait_asynccnt(0);
    asm volatile("" ::: "memory"); // keep LDS reads after the wait
#else
    asm volatile("s_wait_asynccnt 0" ::: "memory");
#endif

    // ---- Per-lane accumulation over 16 consecutive samples ----
    const int s0 = lane * SAMPLES_PER_LANE;
    float px = 0.f, py = 0.f, pz = 0.f;
    if (lane > 0) { // predecessor of this lane's first sample
        px = lx[3 * (s0 - 1) + 0];
        py = lx[3 * (s0 - 1) + 1];
        pz = lx[3 * (s0 - 1) + 2];
    }

    float acc = 0.f;
    #pragma unroll
    for (int j = 0; j < SAMPLES_PER_LANE; ++j) {
        const int s = s0 + j;
        const float cx = lx[3 * s + 0];
        const float cy = lx[3 * s + 1];
        const float cz = lx[3 * s + 2];

        // idx = int32(128 * (x+1) * 0.5) — powers of two are exact, and the
        // C float->int cast truncates toward zero just like jnp astype(int32).
        const int ix = (int)(64.f * (cx + 1.f));
        const int iy = (int)(64.f * (cy + 1.f));
        const int iz = (int)(64.f * (cz + 1.f));
        const bool oob = (ix < 0) | (ix >= GRES) |
                         (iy < 0) | (iy >= GRES) |
                         (iz < 0) | (iz >= GRES);
        const int c0 = min(max(ix, 0), GRES - 1);
        const int c1 = min(max(iy, 0), GRES - 1);
        const int c2 = min(max(iz, 0), GRES - 1);

        float dens = grid[(c0 * GRES + c1) * GRES + c2]; // L2-resident gather
        dens = oob ? 0.f : dens;

        const float dx = cx - px, dy = cy - py, dz = cz - pz;
        const float dist = (s == 0)
            ? 1.0f
            : SCALE * sqrtf(fmaf(dx, dx, fmaf(dy, dy, dz * dz)));

        acc = fmaf(dens, dist, acc);
        px = cx; py = cy; pz = cz;
    }

    // ---- wave32 reduction ----
    #pragma unroll
    for (int off = 16; off > 0; off >>= 1)
        acc += __shfl_xor(acc, off, 32);

    if (lane == 0)
        out[ray] = expf(-acc);
}

extern "C" void kernel_launch(void* const* d_in, const int* in_sizes, int n_in,
                              void* d_out, int out_size, void* d_ws, size_t ws_size,
                              hipStream_t stream)
{
    const float* x    = (const float*)d_in[0]; // (8192, 512, 3) f32
    const float* grid = (const float*)d_in[1]; // (128,128,128) f32
    float*       out  = (float*)d_out;         // (8192, 1) f32

    const int n_rays = in_sizes[0] / FLOATS_PER_RAY;
    const int blocks = (n_rays + WAVES_PER_BLOCK - 1) / WAVES_PER_BLOCK;
    em_grid_kernel<<<blocks, WAVES_PER_BLOCK * LANES, 0, stream>>>(x, grid, out, n_rays);
}